// Net_67370857005644
// MI455X (gfx1250) — compile-verified
//
#include <hip/hip_runtime.h>
#include <hip/hip_bf16.h>
#include <math.h>

// Problem constants (match reference)
#define NNODES 51200
#define NEDGES 819200
#define BATCHG 128
#define NPG    400
#define K1 360
#define K2 324
#define K3 292

typedef __attribute__((ext_vector_type(2))) float v2f;
typedef __attribute__((ext_vector_type(8))) float v8f;

static __device__ __forceinline__ int gtid() {
  return blockIdx.x * blockDim.x + threadIdx.x;
}

// ---------------- elementwise helpers ----------------
__global__ void k_fill(float* p, float v, int n) {
  int i = gtid();
  if (i < n) p[i] = v;
}
__global__ void k_copy(float* o, const float* a, int n) {
  int i = gtid();
  if (i < n) o[i] = a[i];
}

// ---------------- WMMA fp32 GEMM: H = X(MxK) @ W(KxN) (+bias) ----------------
// Block = 8 waves covering a 128x16 output strip (one tile-column).
// The K x 16 slab of W is staged in LDS once per block and shared by all waves;
// A fragments stream from global as coalesced float2 loads.
// Requires: M % 128 == 0, K % 4 == 0, N % 16 == 0, K <= 128.
__global__ void k_gemm_wmma(const float* __restrict__ X, const float* __restrict__ W,
                            const float* __restrict__ bias, float* __restrict__ H,
                            int M, int K, int Nd) {
  __shared__ float Bs[128 * 16];  // max K = 128
  int lane = threadIdx.x & 31;
  int wid  = threadIdx.x >> 5;
  int tilesN = Nd >> 4;
  int mblk = blockIdx.x / tilesN;          // which 128-row strip
  int tn   = blockIdx.x - mblk * tilesN;   // tile column (uniform per block)

  // cooperative load of W slab: Bs[k*16 + n] = W[k][tn*16 + n]
  for (int i = threadIdx.x; i < K * 16; i += 256) {
    int kk = i >> 4, nn = i & 15;
    Bs[i] = W[(size_t)kk * Nd + tn * 16 + nn];
  }
  __syncthreads();

  int tm   = mblk * 8 + wid;               // this wave's 16-row tile
  int l15  = lane & 15;
  int koff = (lane >> 4) << 1;             // lanes 0-15: K=0,1 ; lanes 16-31: K=2,3
  const float* arow = X + (size_t)(tm * 16 + l15) * K;
  int ncol = tn * 16 + l15;

  v8f c = {};
  for (int k = 0; k < K; k += 4) {
    v2f a, b;
    a.x = arow[k + koff];
    a.y = arow[k + koff + 1];
    b.x = Bs[(k + koff) * 16 + l15];
    b.y = Bs[(k + koff + 1) * 16 + l15];
    c = __builtin_amdgcn_wmma_f32_16x16x4_f32(false, a, false, b, (short)0, c,
                                              false, false);
  }
  float bb = bias ? bias[ncol] : 0.0f;
  int rbase = tm * 16 + ((lane >> 4) << 3);  // VGPR v -> M=v (lanes 0-15), M=v+8
#pragma unroll
  for (int v = 0; v < 8; ++v)
    H[(size_t)(rbase + v) * Nd + ncol] = c[v] + bb;
}

// ---------------- GCN pieces ----------------
__global__ void k_edge_deg(const int* __restrict__ dst, const float* __restrict__ emask,
                           float* __restrict__ deg) {
  int e = gtid();
  if (e < NEDGES) atomicAdd(&deg[dst[e]], emask[e]);
}
__global__ void k_dinv(float* deg) {  // in place: deg -> D^-1/2
  int i = gtid();
  if (i < NNODES) {
    float d = deg[i];
    deg[i] = d > 0.0f ? rsqrtf(d) : 0.0f;
  }
}
// one thread per (edge, 4 features): float4 read of h, 4 L2 atomics into agg
__global__ void k_edge_scatter(const float* __restrict__ h, const int* __restrict__ src,
                               const int* __restrict__ dst, const float* __restrict__ dinv,
                               const float* __restrict__ emask, float* __restrict__ agg,
                               int D) {
  int q = D >> 2;
  int idx = gtid();
  int total = NEDGES * q;
  if (idx >= total) return;
  int e = idx / q, c = idx - e * q;
  float w = emask[e];
  if (w == 0.0f) return;
  int s = src[e], t = dst[e];
  float norm = dinv[s] * dinv[t] * w;
  if (norm == 0.0f) return;
  int d = c << 2;
  const float4 hv = *(const float4*)(h + (size_t)s * D + d);
  float* ap = agg + (size_t)t * D + d;
  atomicAdd(ap + 0, hv.x * norm);
  atomicAdd(ap + 1, hv.y * norm);
  atomicAdd(ap + 2, hv.z * norm);
  atomicAdd(ap + 3, hv.w * norm);
}
__global__ void k_combine(const float* __restrict__ agg, const float* __restrict__ h,
                          const float* __restrict__ dinv, const float* __restrict__ nmask,
                          const float* __restrict__ bias, float* __restrict__ xo, int D) {
  int idx = gtid();
  if (idx >= NNODES * D) return;
  int i = idx / D, d = idx - i * D;
  float nm = nmask[i], di = dinv[i];
  float v = (agg[idx] + h[idx] * di * di * nm + bias[d]) * nm;
  xo[idx] = fmaxf(v, 0.0f);
}

// ---------------- TopK pooling ----------------
__global__ void k_pnorm(const float* p, int D, float* pn) {
  __shared__ float s[128];
  int t = threadIdx.x;
  float v = (t < D) ? p[t] : 0.0f;
  s[t] = v * v;
  __syncthreads();
  for (int st = 64; st > 0; st >>= 1) {
    if (t < st) s[t] += s[t + st];
    __syncthreads();
  }
  if (t == 0) pn[0] = sqrtf(s[0]);
}
__global__ void k_score(const float* __restrict__ x, const float* __restrict__ p,
                        const float* __restrict__ pn, const float* __restrict__ nmask,
                        float* __restrict__ score, float* __restrict__ sel, int D) {
  int i = gtid();
  if (i >= NNODES) return;
  float acc = 0.0f;
  for (int d = 0; d < D; ++d) acc += x[(size_t)i * D + d] * p[d];
  float s = acc / pn[0];
  score[i] = s;
  sel[i] = nmask[i] > 0.0f ? s : -INFINITY;
}
// per-graph descending bitonic sort of 512 (400 real + pad), threshold = k-th value
__global__ void k_topk(const float* __restrict__ sel, float* __restrict__ thresh, int K) {
  __shared__ float s[512];
  int g = blockIdx.x, t = threadIdx.x;
  s[t] = (t < NPG) ? sel[g * NPG + t] : -INFINITY;
  __syncthreads();
  for (int k = 2; k <= 512; k <<= 1)
    for (int j = k >> 1; j > 0; j >>= 1) {
      int ixj = t ^ j;
      if (ixj > t) {
        bool up = ((t & k) == 0);
        float a = s[t], b = s[ixj];
        if (up ? (a < b) : (a > b)) { s[t] = b; s[ixj] = a; }
      }
      __syncthreads();
    }
  if (t == 0) thresh[g] = s[K - 1];
}
__global__ void k_pool_node(const float* __restrict__ sel, const float* __restrict__ score,
                            const float* __restrict__ thresh, float* __restrict__ nmask,
                            float* __restrict__ factor) {
  int i = gtid();
  if (i >= NNODES) return;
  int g = i / NPG;
  float nm = (sel[i] >= thresh[g]) ? 1.0f : 0.0f;
  nmask[i] = nm;
  factor[i] = tanhf(score[i]) * nm;
}
__global__ void k_scale(float* __restrict__ x, const float* __restrict__ factor, int D) {
  int idx = gtid();
  if (idx >= NNODES * D) return;
  x[idx] *= factor[idx / D];
}
__global__ void k_emask(const int* __restrict__ src, const int* __restrict__ dst,
                        const float* __restrict__ nmask, float* __restrict__ emask) {
  int e = gtid();
  if (e < NEDGES) emask[e] *= nmask[src[e]] * nmask[dst[e]];
}

// ---------------- readout + MLP head ----------------
__global__ void k_maxpool(const float* __restrict__ x, float* __restrict__ out) {
  int idx = gtid();
  if (idx >= BATCHG * 128) return;
  int g = idx >> 7, d = idx & 127;
  const float* base = x + (size_t)g * NPG * 128 + d;
  float m = -INFINITY;
  for (int n = 0; n < NPG; ++n) m = fmaxf(m, base[(size_t)n * 128]);
  out[idx] = m;
}
__global__ void k_bn_relu(const float* __restrict__ h, float* __restrict__ o,
                          const float* __restrict__ ga, const float* __restrict__ be,
                          int D) {
  __shared__ float s1[128], s2[128];
  int d = blockIdx.x, t = threadIdx.x;  // 128 threads = batch rows
  float v = h[(size_t)t * D + d];
  s1[t] = v;
  s2[t] = v * v;
  __syncthreads();
  for (int st = 64; st > 0; st >>= 1) {
    if (t < st) { s1[t] += s1[t + st]; s2[t] += s2[t + st]; }
    __syncthreads();
  }
  float mean = s1[0] * (1.0f / 128.0f);
  float var  = s2[0] * (1.0f / 128.0f) - mean * mean;  // biased, matches ref
  float out  = ga[d] * (v - mean) * rsqrtf(var + 1e-5f) + be[d];
  o[(size_t)t * D + d] = fmaxf(out, 0.0f);
}
__global__ void k_lin5(const float* __restrict__ h, const float* __restrict__ W,
                       const float* __restrict__ b, float* __restrict__ out) {
  int r = gtid();
  if (r >= BATCHG) return;
  float acc = b[0];
  for (int d = 0; d < 16; ++d) acc += h[r * 16 + d] * W[d];
  out[r] = acc;
}

// ---------------- host orchestration ----------------
static inline int cdiv(int a, int b) { return (a + b - 1) / b; }

extern "C" void kernel_launch(void* const* d_in, const int* in_sizes, int n_in,
                              void* d_out, int out_size, void* d_ws, size_t ws_size,
                              hipStream_t stream) {
  (void)in_sizes; (void)n_in; (void)out_size; (void)ws_size;
  const float* x0  = (const float*)d_in[0];
  const int*   ei  = (const int*)d_in[1];
  const int*   src = ei;
  const int*   dst = ei + NEDGES;
  const float* W[4]  = {(const float*)d_in[3], (const float*)d_in[5],
                        (const float*)d_in[7], (const float*)d_in[9]};
  const float* bW[4] = {(const float*)d_in[4], (const float*)d_in[6],
                        (const float*)d_in[8], (const float*)d_in[10]};
  const float* P[3]  = {(const float*)d_in[11], (const float*)d_in[12],
                        (const float*)d_in[13]};
  const float* Lw[5] = {(const float*)d_in[14], (const float*)d_in[16],
                        (const float*)d_in[18], (const float*)d_in[20],
                        (const float*)d_in[22]};
  const float* Lb[5] = {(const float*)d_in[15], (const float*)d_in[17],
                        (const float*)d_in[19], (const float*)d_in[21],
                        (const float*)d_in[23]};
  const float* G[4]  = {(const float*)d_in[24], (const float*)d_in[26],
                        (const float*)d_in[28], (const float*)d_in[30]};
  const float* BE[4] = {(const float*)d_in[25], (const float*)d_in[27],
                        (const float*)d_in[29], (const float*)d_in[31]};
  float* out = (float*)d_out;

  // workspace carve-up
  float* ws     = (float*)d_ws;
  float* xa     = ws;                          // N*128 current features
  float* xb     = xa + (size_t)NNODES * 128;   // N*128 h = X@W
  float* agg    = xb + (size_t)NNODES * 128;   // N*128 segment sums
  float* deg    = agg + (size_t)NNODES * 128;  // N   (becomes dinv)
  float* score  = deg + NNODES;                // N
  float* sel    = score + NNODES;              // N
  float* factor = sel + NNODES;                // N
  float* nmask  = factor + NNODES;             // N
  float* emask  = nmask + NNODES;              // E
  float* thresh = emask + NEDGES;              // B
  float* pn     = thresh + BATCHG;             // 4
  float* hg     = pn + 4;                      // B*128 pooled
  float* m1     = hg + (size_t)BATCHG * 128;   // B*128
  float* m2     = m1 + (size_t)BATCHG * 128;   // B*128

  const int TB = 256;
  k_fill<<<cdiv(NNODES, TB), TB, 0, stream>>>(nmask, 1.0f, NNODES);
  k_fill<<<cdiv(NEDGES, TB), TB, 0, stream>>>(emask, 1.0f, NEDGES);

  const int Din[4]  = {64, 32, 64, 128};
  const int Dout[4] = {32, 64, 128, 128};
  const int Ks[3]   = {K1, K2, K3};

  for (int L = 0; L < 4; ++L) {
    const float* xin = (L == 0) ? x0 : xa;
    int Di = Din[L], Do = Dout[L];
    // h = X @ W  (WMMA fp32, LDS-staged B slab, 128x16 strip per block)
    int blocks = (NNODES / 128) * (Do / 16);
    k_gemm_wmma<<<blocks, 256, 0, stream>>>(xin, W[L], nullptr, xb,
                                            NNODES, Di, Do);
    // degrees -> D^-1/2
    k_copy<<<cdiv(NNODES, TB), TB, 0, stream>>>(deg, nmask, NNODES);
    k_edge_deg<<<cdiv(NEDGES, TB), TB, 0, stream>>>(dst, emask, deg);
    k_dinv<<<cdiv(NNODES, TB), TB, 0, stream>>>(deg);
    // agg = segment_sum(h[src]*norm, dst)
    k_fill<<<cdiv(NNODES * Do, TB), TB, 0, stream>>>(agg, 0.0f, NNODES * Do);
    k_edge_scatter<<<cdiv(NEDGES * (Do / 4), TB), TB, 0, stream>>>(
        xb, src, dst, deg, emask, agg, Do);
    // x = relu((agg + self + b) * nmask)
    k_combine<<<cdiv(NNODES * Do, TB), TB, 0, stream>>>(agg, xb, deg, nmask,
                                                        bW[L], xa, Do);
    if (L < 3) {
      k_pnorm<<<1, 128, 0, stream>>>(P[L], Do, pn);
      k_score<<<cdiv(NNODES, TB), TB, 0, stream>>>(xa, P[L], pn, nmask, score,
                                                   sel, Do);
      k_topk<<<BATCHG, 512, 0, stream>>>(sel, thresh, Ks[L]);
      k_pool_node<<<cdiv(NNODES, TB), TB, 0, stream>>>(sel, score, thresh,
                                                       nmask, factor);
      k_scale<<<cdiv(NNODES * Do, TB), TB, 0, stream>>>(xa, factor, Do);
      k_emask<<<cdiv(NEDGES, TB), TB, 0, stream>>>(src, dst, nmask, emask);
    }
  }

  // global max pool -> (B, 128)
  k_maxpool<<<cdiv(BATCHG * 128, TB), TB, 0, stream>>>(xa, hg);

  // MLP head: (Lw,Lb) + BN + relu, via WMMA GEMM (M=128 -> one strip)
  const int mK[4] = {128, 64, 64, 32};
  const int mN[4] = {64, 64, 32, 16};
  const float* hin = hg;
  for (int L = 0; L < 4; ++L) {
    int blocks = (BATCHG / 128) * (mN[L] / 16);
    k_gemm_wmma<<<blocks, 256, 0, stream>>>(hin, Lw[L], Lb[L], m1,
                                            BATCHG, mK[L], mN[L]);
    k_bn_relu<<<mN[L], 128, 0, stream>>>(m1, m2, G[L], BE[L], mN[L]);
    hin = m2;
  }
  k_lin5<<<cdiv(BATCHG, 128), 128, 0, stream>>>(m2, Lw[4], Lb[4], out);
}